// LIFModFeedForwardCell_76124000354679
// MI455X (gfx1250) — compile-verified
//
#include <hip/hip_runtime.h>
#include <hip/hip_bf16.h>

// LIF-with-refractory constants (compile-time folded from the reference)
#define K_MEM    0.1f   // DT * TAU_MEM_INV
#define K_SYN    0.2f   // DT * TAU_SYN_INV
#define K_REFRAC 1.0f   // DT * TAU_REFRAC_INV
#define V_TH     1.0f
#define RHO_RST  5.0f
// V_LEAK = V_RESET = 0 -> folded away

typedef __attribute__((ext_vector_type(4))) float v4f;
typedef __attribute__((ext_vector_type(4))) int   v4i;

// ---------------------------------------------------------------------------
// Vectorized path: one float4 per lane. All contiguous streams go through
// 128-bit VMEM with TH=NT (single-use streaming data; keep it out of WGP$/L2
// so the 512MB delay buffer's random gathers get whatever cache is useful).
// The per-neuron delay gather is inherently uncoalesced (random t in [0,64),
// plane stride 8MB) -> scalar 32-bit NT loads, branchless select for delay==0.
// ---------------------------------------------------------------------------
__global__ __launch_bounds__(256) void lif_step_vec4(
    const v4f* __restrict__ i_new4,
    const v4f* __restrict__ v4,
    const v4f* __restrict__ i4,
    const v4f* __restrict__ rho4,
    const float* __restrict__ buffer,   // [T, N] flat
    const v4i* __restrict__ delay4,
    float* __restrict__ out,            // [5, N] flat
    int n)                              // N, divisible by 4
{
    const int n4 = n >> 2;
    const int t  = blockIdx.x * blockDim.x + threadIdx.x;
    if (t >= n4) return;

    const v4f inw = __builtin_nontemporal_load(i_new4 + t);
    const v4f vv  = __builtin_nontemporal_load(v4     + t);
    const v4f ii  = __builtin_nontemporal_load(i4     + t);
    const v4f rr  = __builtin_nontemporal_load(rho4   + t);
    const v4i dd  = __builtin_nontemporal_load(delay4 + t);

    v4f z, vn, idec, rn, zd;
#pragma unroll
    for (int c = 0; c < 4; ++c) {
        const float refrac = (rr[c] > 0.0f) ? 1.0f : 0.0f;            // Theta(rho)
        const float v_dec  = vv[c] + K_MEM * (1.0f - refrac) * (ii[c] - vv[c]);
        const float i_dec  = ii[c] - K_SYN * ii[c] + inw[c];
        const float zz     = (v_dec > V_TH) ? 1.0f : 0.0f;
        const float v_new  = (1.0f - zz) * v_dec;                     // V_RESET = 0
        const float r_new  = rr[c] - K_REFRAC * refrac + zz * RHO_RST;

        // buffer_new = [z ; buffer[:-1]]; gather at delay d:
        //   d == 0 -> z (just pushed), d > 0 -> buffer[d-1]
        const int    del  = dd[c];
        const int    dm1  = (del > 0) ? (del - 1) : 0;                // clamp: load stays in-bounds
        const size_t goff = (size_t)dm1 * (size_t)n + (size_t)(4 * t + c);
        const float  bval = __builtin_nontemporal_load(buffer + goff);
        const float  z_del = (del == 0) ? zz : bval;                  // branchless select

        z[c]    = zz;
        vn[c]   = v_new;
        idec[c] = i_dec;
        rn[c]   = r_new;
        zd[c]   = z_del;
    }

    // out planes: [z_delayed, z, v_new, i_dec, rho_new], each N floats.
    // n % 4 == 0 -> every plane base is 16B aligned -> b128 stores.
    v4f* out4 = (v4f*)out;
    __builtin_nontemporal_store(zd,   out4 + 0 * (size_t)n4 + t);
    __builtin_nontemporal_store(z,    out4 + 1 * (size_t)n4 + t);
    __builtin_nontemporal_store(vn,   out4 + 2 * (size_t)n4 + t);
    __builtin_nontemporal_store(idec, out4 + 3 * (size_t)n4 + t);
    __builtin_nontemporal_store(rn,   out4 + 4 * (size_t)n4 + t);
}

// ---------------------------------------------------------------------------
// Scalar fallback (only used if N % 4 != 0; never for the reference shapes,
// but keeps kernel_launch total).
// ---------------------------------------------------------------------------
__global__ __launch_bounds__(256) void lif_step_scalar(
    const float* __restrict__ i_new,
    const float* __restrict__ v,
    const float* __restrict__ i,
    const float* __restrict__ rho,
    const float* __restrict__ buffer,
    const int*  __restrict__ delay,
    float* __restrict__ out,
    int n)
{
    const int e = blockIdx.x * blockDim.x + threadIdx.x;
    if (e >= n) return;

    const float rr = rho[e];
    const float vv = v[e];
    const float ii = i[e];

    const float refrac = (rr > 0.0f) ? 1.0f : 0.0f;
    const float v_dec  = vv + K_MEM * (1.0f - refrac) * (ii - vv);
    const float i_dec  = ii - K_SYN * ii + i_new[e];
    const float zz     = (v_dec > V_TH) ? 1.0f : 0.0f;
    const float v_new  = (1.0f - zz) * v_dec;
    const float r_new  = rr - K_REFRAC * refrac + zz * RHO_RST;

    const int   del  = delay[e];
    const int   dm1  = (del > 0) ? (del - 1) : 0;
    const float bval = buffer[(size_t)dm1 * (size_t)n + (size_t)e];
    const float z_del = (del == 0) ? zz : bval;

    out[0 * (size_t)n + e] = z_del;
    out[1 * (size_t)n + e] = zz;
    out[2 * (size_t)n + e] = v_new;
    out[3 * (size_t)n + e] = i_dec;
    out[4 * (size_t)n + e] = r_new;
}

extern "C" void kernel_launch(void* const* d_in, const int* in_sizes, int n_in,
                              void* d_out, int out_size, void* d_ws, size_t ws_size,
                              hipStream_t stream) {
    // setup_inputs() order: i_new, v, i, rho, buffer, delay_idx
    const float* i_new  = (const float*)d_in[0];
    const float* v      = (const float*)d_in[1];
    const float* i      = (const float*)d_in[2];
    const float* rho    = (const float*)d_in[3];
    const float* buffer = (const float*)d_in[4];
    const int*   delay  = (const int*)  d_in[5];
    float*       out    = (float*)d_out;

    const int n = in_sizes[0];          // B*C*H*W = 2,097,152

    const int block = 256;              // 8 wave32 per workgroup
    if ((n & 3) == 0) {
        const int n4   = n >> 2;
        const int grid = (n4 + block - 1) / block;   // 2048 blocks
        lif_step_vec4<<<grid, block, 0, stream>>>(
            (const v4f*)i_new, (const v4f*)v, (const v4f*)i, (const v4f*)rho,
            buffer, (const v4i*)delay, out, n);
    } else {
        const int grid = (n + block - 1) / block;
        lif_step_scalar<<<grid, block, 0, stream>>>(
            i_new, v, i, rho, buffer, delay, out, n);
    }
}